// Lattice_18794776887559
// MI455X (gfx1250) — compile-verified
//
#include <hip/hip_runtime.h>
#include <math.h>

#define SPACING 32
#define NPTS    2048
#define NB      8
#define GRID_M  (SPACING*SPACING*SPACING)   // 32768
#define BLK     256                          // 8 wave32s per block
#define NWAVES  (BLK/32)
#define CHUNK   16
#define NCHUNKS (NPTS/CHUNK)
#define SCROW   20                           // score row stride (floats): 16B-aligned rows
#define TWO_PI_F 6.283185307179586f

typedef __attribute__((ext_vector_type(2))) float v2f;
typedef __attribute__((ext_vector_type(8))) float v8f;

// ---------------------------------------------------------------------------
// Kernel 1: brute-force KNN (top-8) + inverse-distance^2 interpolation.
// One grid point per lane. Distances via V_WMMA_F32_16X16X4_F32:
//   A row (16x4)  = (-2gx, -2gy, -2gz, 1)
//   B col (4x16)  = ( px,   py,   pz,  |p|^2)
//   score = A*B   = -2 g.p + |p|^2  == d^2 - |g|^2 (row-constant), so the
//   per-row top-k ordering is exactly that of d^2.
// Score tiles are exchanged through a wave-private LDS region; same-wave LDS
// ops complete in order (DScnt), so no block barriers are needed in the loop.
// ---------------------------------------------------------------------------
__global__ __launch_bounds__(BLK) void knn_interp_kernel(
    const float* __restrict__ flow,         // (B, NPTS, 3)
    const float* __restrict__ coords,       // (B, NPTS, 3)
    const float* __restrict__ grid_coords,  // (B, GRID_M, 3)
    float* __restrict__ gf)                 // (B, GRID_M, 3) out
{
    // Staging buffer (24KB): first used for async AoS coords staging, then
    // reused as the per-wave score-tile buffer (needs 8*32*20 = 5120 floats).
    __shared__ __align__(16) float s_stage[NPTS * 3];
    __shared__ float s_px[NPTS];
    __shared__ float s_py[NPTS];
    __shared__ float s_pz[NPTS];
    __shared__ float s_p2[NPTS];

    const int b     = blockIdx.y;
    const int mbase = blockIdx.x * BLK;
    const int tid   = threadIdx.x;
    const int lane  = tid & 31;
    const int wave  = tid >> 5;
    const int half  = (lane < 16) ? 0 : 1;   // which 16-lane half of the wave
    const int l16   = lane & 15;

    // ---- Async-copy all candidate coords (AoS) into LDS: 1536 x 16B ----
    {
        const float* src = coords + (size_t)b * NPTS * 3;
        unsigned lds0 = (unsigned)(size_t)&s_stage[0];   // low 32 bits = LDS offset
#pragma unroll
        for (int i = 0; i < (NPTS * 3 / 4) / BLK; ++i) { // 6 iterations
            int e = i * BLK + tid;                       // float4 index
            unsigned lds = lds0 + (unsigned)e * 16u;
            unsigned long long ga = (unsigned long long)(size_t)(src + (size_t)e * 4);
            asm volatile("global_load_async_to_lds_b128 %0, %1, off"
                         :: "v"(lds), "v"(ga) : "memory");
        }
        asm volatile("s_wait_asynccnt 0x0" ::: "memory");
    }
    __syncthreads();

    // Scatter AoS -> SoA and precompute |p|^2.
    for (int j = tid; j < NPTS; j += BLK) {
        float x = s_stage[j * 3 + 0];
        float y = s_stage[j * 3 + 1];
        float z = s_stage[j * 3 + 2];
        s_px[j] = x; s_py[j] = y; s_pz[j] = z;
        s_p2[j] = x * x + y * y + z * z;
    }

    // This lane's own grid point (for final exact distances).
    const int m = mbase + tid;
    const float gx = grid_coords[(size_t)(b * GRID_M + m) * 3 + 0];
    const float gy = grid_coords[(size_t)(b * GRID_M + m) * 3 + 1];
    const float gz = grid_coords[(size_t)(b * GRID_M + m) * 3 + 2];

    // Build the two A fragments (rows wave*32 + t*16 + l16) once.
    // A(16x4) layout: lanes 0-15 hold K=0,1 in v0,v1; lanes 16-31 hold K=2,3.
    v2f afrag[2];
#pragma unroll
    for (int t = 0; t < 2; ++t) {
        int row = mbase + wave * 32 + t * 16 + l16;
        float rx = grid_coords[(size_t)(b * GRID_M + row) * 3 + 0];
        float ry = grid_coords[(size_t)(b * GRID_M + row) * 3 + 1];
        float rz = grid_coords[(size_t)(b * GRID_M + row) * 3 + 2];
        if (half == 0) { afrag[t].x = -2.0f * rx; afrag[t].y = -2.0f * ry; }
        else           { afrag[t].x = -2.0f * rz; afrag[t].y = 1.0f;       }
    }

    __syncthreads();  // staging reads done -> s_stage reusable as score tiles

    // Sorted (ascending) running top-8 in registers.
    float bs[NB];
    int   bi[NB];
#pragma unroll
    for (int q = 0; q < NB; ++q) { bs[q] = __builtin_inff(); bi[q] = 0; }

    float* wtile = s_stage + (size_t)(wave * 32) * SCROW;  // wave-private region

    for (int c = 0; c < NCHUNKS; ++c) {
        // B fragment for candidates [c*16, c*16+16).
        // B(4x16) layout: v0 holds rows K=0 (lanes 0-15) / K=2 (lanes 16-31),
        //                 v1 holds rows K=1 / K=3.
        const int n = c * CHUNK + l16;
        v2f bfrag;
        if (half == 0) { bfrag.x = s_px[n]; bfrag.y = s_py[n]; }
        else           { bfrag.x = s_pz[n]; bfrag.y = s_p2[n]; }

#pragma unroll
        for (int t = 0; t < 2; ++t) {
            v8f acc = {0.f, 0.f, 0.f, 0.f, 0.f, 0.f, 0.f, 0.f};
            acc = __builtin_amdgcn_wmma_f32_16x16x4_f32(
                /*neg_a=*/false, afrag[t], /*neg_b=*/false, bfrag,
                /*c_mod=*/(short)0, acc, /*reuse_a=*/false, /*reuse_b=*/false);
            // C/D layout: VGPR i -> row M = i + 8*half, col N = l16.
#pragma unroll
            for (int i = 0; i < 8; ++i)
                wtile[(size_t)(t * 16 + i + 8 * half) * SCROW + l16] = acc[i];
        }
        __builtin_amdgcn_wave_barrier();   // keep DS order; same-wave LDS is in-order

        // Vectorized row read: 4 x ds_load_b128 (rows are 16B-aligned).
        const float4* row = (const float4*)(s_stage + (size_t)(wave * 32 + lane) * SCROW);
        float4 r0 = row[0], r1 = row[1], r2 = row[2], r3 = row[3];

        // Chunk-level early-out: min of the 16 scores vs current 8th best.
        float mn = fminf(fminf(fminf(fminf(r0.x, r0.y), fminf(r0.z, r0.w)),
                               fminf(fminf(r1.x, r1.y), fminf(r1.z, r1.w))),
                         fminf(fminf(fminf(r2.x, r2.y), fminf(r2.z, r2.w)),
                               fminf(fminf(r3.x, r3.y), fminf(r3.z, r3.w))));
        if (mn < bs[NB - 1]) {
            const float vals[16] = {r0.x, r0.y, r0.z, r0.w,
                                    r1.x, r1.y, r1.z, r1.w,
                                    r2.x, r2.y, r2.z, r2.w,
                                    r3.x, r3.y, r3.z, r3.w};
#pragma unroll
            for (int k = 0; k < CHUNK; ++k) {
                float sc = vals[k];
                if (sc < bs[NB - 1]) {
                    float cs = sc;
                    int   ci = c * CHUNK + k;
#pragma unroll
                    for (int q = 0; q < NB; ++q) {
                        bool lt = cs < bs[q];
                        float ts = lt ? bs[q] : cs;
                        int   ti = lt ? bi[q] : ci;
                        bs[q] = lt ? cs : bs[q];
                        bi[q] = lt ? ci : bi[q];
                        cs = ts; ci = ti;
                    }
                }
            }
        }
        __builtin_amdgcn_wave_barrier();   // next chunk's stores stay after reads
    }

    // Inverse-distance^2 interpolation over the 8 selected neighbors.
    float wsum = 0.f, ax = 0.f, ay = 0.f, az = 0.f;
#pragma unroll
    for (int q = 0; q < NB; ++q) {
        int j = bi[q];
        float dx = gx - s_px[j];
        float dy = gy - s_py[j];
        float dz = gz - s_pz[j];
        float dist = sqrtf(fmaxf(dx * dx + dy * dy + dz * dz, 0.f));
        float t = dist + 1e-8f;
        float w = 1.0f / (t * t);
        wsum += w;
        const float* f = &flow[(size_t)(b * NPTS + j) * 3];
        ax += w * f[0];
        ay += w * f[1];
        az += w * f[2];
    }
    float inv = 1.0f / wsum;
    gf[(size_t)(b * GRID_M + m) * 3 + 0] = ax * inv;
    gf[(size_t)(b * GRID_M + m) * 3 + 1] = ay * inv;
    gf[(size_t)(b * GRID_M + m) * 3 + 2] = az * inv;
}

// ---------------------------------------------------------------------------
// Kernel 2: divergence (torch.gradient semantics) + per-block |.| sums.
// ---------------------------------------------------------------------------
__global__ __launch_bounds__(256) void div_kernel(
    const float* __restrict__ gf, float* __restrict__ partials)
{
    const int b = blockIdx.y;
    const int m = blockIdx.x * 256 + threadIdx.x;
    const int ix = m >> 10;
    const int iy = (m >> 5) & 31;
    const int iz = m & 31;
    const float H = TWO_PI_F / (float)SPACING;

    auto F = [&](int x, int y, int z, int c) -> float {
        return gf[(size_t)((b * GRID_M) + (x << 10) + (y << 5) + z) * 3 + c];
    };

    float dx = (ix == 0)  ? (F(1, iy, iz, 0)  - F(0, iy, iz, 0))  / H
             : (ix == 31) ? (F(31, iy, iz, 0) - F(30, iy, iz, 0)) / H
                          : (F(ix + 1, iy, iz, 0) - F(ix - 1, iy, iz, 0)) / (2.0f * H);
    float dy = (iy == 0)  ? (F(ix, 1, iz, 1)  - F(ix, 0, iz, 1))  / H
             : (iy == 31) ? (F(ix, 31, iz, 1) - F(ix, 30, iz, 1)) / H
                          : (F(ix, iy + 1, iz, 1) - F(ix, iy - 1, iz, 1)) / (2.0f * H);
    float dz = (iz == 0)  ? (F(ix, iy, 1, 2)  - F(ix, iy, 0, 2))  / H
             : (iz == 31) ? (F(ix, iy, 31, 2) - F(ix, iy, 30, 2)) / H
                          : (F(ix, iy, iz + 1, 2) - F(ix, iy, iz - 1, 2)) / (2.0f * H);

    __shared__ float red[256];
    red[threadIdx.x] = fabsf(dx + dy + dz);
    __syncthreads();
    for (int s = 128; s > 0; s >>= 1) {
        if (threadIdx.x < s) red[threadIdx.x] += red[threadIdx.x + s];
        __syncthreads();
    }
    if (threadIdx.x == 0)
        partials[b * (GRID_M / 256) + blockIdx.x] = red[0];
}

// ---------------------------------------------------------------------------
// Kernel 3: deterministic final reduction of 256 partials -> scalar mean.
// ---------------------------------------------------------------------------
__global__ __launch_bounds__(256) void finalize_kernel(
    const float* __restrict__ partials, float* __restrict__ out)
{
    __shared__ float red[256];
    red[threadIdx.x] = partials[threadIdx.x];
    __syncthreads();
    for (int s = 128; s > 0; s >>= 1) {
        if (threadIdx.x < s) red[threadIdx.x] += red[threadIdx.x + s];
        __syncthreads();
    }
    if (threadIdx.x == 0)
        out[0] = red[0] / (float)(2 * GRID_M);
}

// ---------------------------------------------------------------------------
extern "C" void kernel_launch(void* const* d_in, const int* in_sizes, int n_in,
                              void* d_out, int out_size, void* d_ws, size_t ws_size,
                              hipStream_t stream) {
    (void)in_sizes; (void)n_in; (void)out_size; (void)ws_size;
    const float* flow        = (const float*)d_in[0];  // (2, 2048, 3)
    const float* coords      = (const float*)d_in[1];  // (2, 2048, 3)
    const float* grid_coords = (const float*)d_in[2];  // (2, 32768, 3)
    float* out = (float*)d_out;                        // scalar

    float* gf       = (float*)d_ws;                    // (2, 32768, 3)
    float* partials = gf + (size_t)2 * GRID_M * 3;     // 2 * 128 = 256 floats

    dim3 g1(GRID_M / BLK, 2);
    knn_interp_kernel<<<g1, BLK, 0, stream>>>(flow, coords, grid_coords, gf);

    dim3 g2(GRID_M / 256, 2);
    div_kernel<<<g2, 256, 0, stream>>>(gf, partials);

    finalize_kernel<<<1, 256, 0, stream>>>(partials, out);
}